// NasPhy10000Cell_36816459661690
// MI455X (gfx1250) — compile-verified
//
#include <hip/hip_runtime.h>
#include <hip/hip_bf16.h>
#include <math.h>

typedef float v2f __attribute__((ext_vector_type(2)));
typedef float v8f __attribute__((ext_vector_type(8)));

#define N_NODES   100000
#define N_EDGES   1200000
#define CUR_DIM   256
#define HIDDEN_DIM 256
#define OUT_DIM   64

// LDS row strides padded to avoid 16-way bank conflicts (stride % 64 == 4)
#define HS_STRIDE  260
#define H1S_STRIDE 68

// ---------------------------------------------------------------------------
// Fused GEMM kernel: one wave (32 threads) computes a 16-row tile.
//   h   = x @ W_pre + b_pre                  (16 x 256, kept in LDS)
//   h1  = leaky_relu(h @ W_lin + b_lin)      (16 x 64, to global + LDS)
//   xw  = h1 @ W_init                        (16 x 64, to global)
//   hr  = h1 @ W_root                        (16 x 64, to global)
// All matmuls via V_WMMA_F32_16X16X4_F32 (exact f32 semantics).
// ---------------------------------------------------------------------------
__global__ __launch_bounds__(32) void fused_gemm_kernel(
    const float* __restrict__ x,      // [N,256]
    const float* __restrict__ Wpre,   // [256,256]
    const float* __restrict__ bpre,   // [256]
    const float* __restrict__ Wlin,   // [256,64]
    const float* __restrict__ blin,   // [64]
    const float* __restrict__ Winit,  // [64,64]
    const float* __restrict__ Wroot,  // [64,64]
    float* __restrict__ h1,           // [N,64]
    float* __restrict__ xw,           // [N,64]
    float* __restrict__ hr)           // [N,64]
{
    __shared__ float hs[16 * HS_STRIDE];    // h tile  (16 x 256, padded)
    __shared__ float h1s[16 * H1S_STRIDE];  // h1 tile (16 x 64, padded)

    const int lane = threadIdx.x;       // 0..31
    const int mrow = lane & 15;         // M index / N index within 16-tile
    const int hi   = lane >> 4;         // K half selector (0: K=0,1  1: K=2,3)
    const long long row0 = (long long)blockIdx.x * 16;

    // ---- Stage 1: h = x @ W_pre + b_pre -> hs -------------------------------
    const float* xrow = x + (row0 + mrow) * CUR_DIM;
    for (int nb = 0; nb < HIDDEN_DIM / 16; ++nb) {
        const int n0 = nb * 16;
        const float bias = bpre[n0 + mrow];
        v8f acc;
        #pragma unroll
        for (int r = 0; r < 8; ++r) acc[r] = bias;

        #pragma unroll 4
        for (int k0 = 0; k0 < CUR_DIM; k0 += 4) {
            // A fragment: x[row0+mrow, k0 + 2*hi + {0,1}]
            v2f a = *(const v2f*)(xrow + k0 + 2 * hi);
            // B fragment: W_pre[k0 + 2*hi + {0,1}, n0 + mrow]
            v2f b;
            b.x = Wpre[(k0 + 2 * hi)     * HIDDEN_DIM + n0 + mrow];
            b.y = Wpre[(k0 + 2 * hi + 1) * HIDDEN_DIM + n0 + mrow];
            acc = __builtin_amdgcn_wmma_f32_16x16x4_f32(
                false, a, false, b, (short)0, acc, false, false);
        }
        // C/D layout: VGPR r, lanes 0-15 -> M=r, lanes 16-31 -> M=r+8; N=mrow
        #pragma unroll
        for (int r = 0; r < 8; ++r)
            hs[(r + 8 * hi) * HS_STRIDE + n0 + mrow] = acc[r];
    }
    __syncthreads();

    // ---- Stage 2: h1 = leaky_relu(hs @ W_lin + b_lin) -----------------------
    for (int nb = 0; nb < OUT_DIM / 16; ++nb) {
        const int n0 = nb * 16;
        const float bias = blin[n0 + mrow];
        v8f acc;
        #pragma unroll
        for (int r = 0; r < 8; ++r) acc[r] = bias;

        #pragma unroll 4
        for (int k0 = 0; k0 < HIDDEN_DIM; k0 += 4) {
            v2f a;
            a.x = hs[mrow * HS_STRIDE + k0 + 2 * hi];
            a.y = hs[mrow * HS_STRIDE + k0 + 2 * hi + 1];
            v2f b;
            b.x = Wlin[(k0 + 2 * hi)     * OUT_DIM + n0 + mrow];
            b.y = Wlin[(k0 + 2 * hi + 1) * OUT_DIM + n0 + mrow];
            acc = __builtin_amdgcn_wmma_f32_16x16x4_f32(
                false, a, false, b, (short)0, acc, false, false);
        }
        #pragma unroll
        for (int r = 0; r < 8; ++r) {
            float v = acc[r];
            v = (v > 0.0f) ? v : 0.01f * v;  // jax.nn.leaky_relu default slope
            const int m = r + 8 * hi;
            h1s[m * H1S_STRIDE + n0 + mrow] = v;
            h1[(row0 + m) * OUT_DIM + n0 + mrow] = v;
        }
    }
    __syncthreads();

    // ---- Stage 3: xw = h1 @ W_init ; hr = h1 @ W_root -----------------------
    for (int nb = 0; nb < OUT_DIM / 16; ++nb) {
        const int n0 = nb * 16;
        v8f acc1, acc2;
        #pragma unroll
        for (int r = 0; r < 8; ++r) { acc1[r] = 0.0f; acc2[r] = 0.0f; }

        #pragma unroll 4
        for (int k0 = 0; k0 < OUT_DIM; k0 += 4) {
            v2f a;
            a.x = h1s[mrow * H1S_STRIDE + k0 + 2 * hi];
            a.y = h1s[mrow * H1S_STRIDE + k0 + 2 * hi + 1];
            v2f b1, b2;
            b1.x = Winit[(k0 + 2 * hi)     * OUT_DIM + n0 + mrow];
            b1.y = Winit[(k0 + 2 * hi + 1) * OUT_DIM + n0 + mrow];
            b2.x = Wroot[(k0 + 2 * hi)     * OUT_DIM + n0 + mrow];
            b2.y = Wroot[(k0 + 2 * hi + 1) * OUT_DIM + n0 + mrow];
            acc1 = __builtin_amdgcn_wmma_f32_16x16x4_f32(
                false, a, false, b1, (short)0, acc1, false, false);
            acc2 = __builtin_amdgcn_wmma_f32_16x16x4_f32(
                false, a, false, b2, (short)0, acc2, false, false);
        }
        #pragma unroll
        for (int r = 0; r < 8; ++r) {
            const int m = r + 8 * hi;
            xw[(row0 + m) * OUT_DIM + n0 + mrow] = acc1[r];
            hr[(row0 + m) * OUT_DIM + n0 + mrow] = acc2[r];
        }
    }
}

// ---------------------------------------------------------------------------
__global__ __launch_bounds__(256) void zero_kernel(float* __restrict__ p, int n) {
    const int i = blockIdx.x * blockDim.x + threadIdx.x;
    if (i < n) p[i] = 0.0f;
}

// deg[dst] += w  (scatter over edges)
__global__ __launch_bounds__(256) void deg_kernel(
    const long long* __restrict__ ei, const float* __restrict__ ew,
    float* __restrict__ deg, int E)
{
    const int e = blockIdx.x * blockDim.x + threadIdx.x;
    if (e < E) {
        const int d = (int)ei[(long long)E + e];  // edge_index[1]
        atomicAdd(&deg[d], ew[e]);
    }
}

// deg -> dis = deg > 0 ? rsqrt(max(deg,1e-30)) : 0   (in place)
__global__ __launch_bounds__(256) void dis_kernel(float* __restrict__ deg, int n) {
    const int i = blockIdx.x * blockDim.x + threadIdx.x;
    if (i < n) {
        const float d = deg[i];
        deg[i] = (d > 0.0f) ? __frsqrt_rn(fmaxf(d, 1e-30f)) : 0.0f;
    }
}

// agg[dst] += norm * xw[src]  — one wave per edge, lane covers a float2 of 64
__global__ __launch_bounds__(256) void scatter_kernel(
    const long long* __restrict__ ei, const float* __restrict__ ew,
    const float* __restrict__ dis, const float* __restrict__ xw,
    float* __restrict__ agg, int E)
{
    const int warp = (int)((blockIdx.x * blockDim.x + threadIdx.x) >> 5);
    const int lane = threadIdx.x & 31;
    if (warp >= E) return;
    const long long s = ei[warp];
    const long long d = ei[(long long)E + warp];
    const float norm = dis[s] * ew[warp] * dis[d];
    const v2f v = *(const v2f*)(xw + s * OUT_DIM + 2 * lane);
    float* dst = agg + d * OUT_DIM + 2 * lane;
    atomicAdd(dst,     norm * v.x);
    atomicAdd(dst + 1, norm * v.y);
}

// out = tanh(h1 + relu(agg + hr + b_arma))   [leaky_relu(relu(z)) == relu(z)]
__global__ __launch_bounds__(256) void final_kernel(
    const float* __restrict__ h1, const float* __restrict__ hr,
    const float* __restrict__ agg, const float* __restrict__ barma,
    float* __restrict__ out, int total)
{
    const int i = blockIdx.x * blockDim.x + threadIdx.x;
    if (i < total) {
        const int f = i & (OUT_DIM - 1);
        const float arma = fmaxf(agg[i] + hr[i] + barma[f], 0.0f);
        out[i] = tanhf(h1[i] + arma);
    }
}

// ---------------------------------------------------------------------------
extern "C" void kernel_launch(void* const* d_in, const int* in_sizes, int n_in,
                              void* d_out, int out_size, void* d_ws, size_t ws_size,
                              hipStream_t stream) {
    const float*     x     = (const float*)d_in[0];
    const long long* ei    = (const long long*)d_in[1];  // int64 [2, E]
    const float*     ew    = (const float*)d_in[2];
    const float*     Wpre  = (const float*)d_in[3];
    const float*     bpre  = (const float*)d_in[4];
    const float*     Wlin  = (const float*)d_in[5];
    const float*     blin  = (const float*)d_in[6];
    const float*     Winit = (const float*)d_in[7];
    const float*     Wroot = (const float*)d_in[8];
    const float*     barma = (const float*)d_in[9];
    float* out = (float*)d_out;

    // Workspace layout (floats): h1 | xw | hr | agg | deg
    const size_t NF = (size_t)N_NODES * OUT_DIM;
    float* ws  = (float*)d_ws;
    float* h1  = ws;
    float* xw  = h1 + NF;
    float* hr  = xw + NF;
    float* agg = hr + NF;
    float* deg = agg + NF;   // agg and deg contiguous -> one zero pass

    // 1) zero agg (N*64) + deg (N) in one launch
    {
        const int n = (int)(NF + N_NODES);
        zero_kernel<<<(n + 255) / 256, 256, 0, stream>>>(agg, n);
    }
    // 2) fused WMMA GEMM chain: h1, xw, hr
    fused_gemm_kernel<<<N_NODES / 16, 32, 0, stream>>>(
        x, Wpre, bpre, Wlin, blin, Winit, Wroot, h1, xw, hr);
    // 3) degree accumulation over edges
    deg_kernel<<<(N_EDGES + 255) / 256, 256, 0, stream>>>(ei, ew, deg, N_EDGES);
    // 4) deg -> D^{-1/2}
    dis_kernel<<<(N_NODES + 255) / 256, 256, 0, stream>>>(deg, N_NODES);
    // 5) edge scatter: agg[dst] += norm * xw[src]  (one wave per edge)
    scatter_kernel<<<(N_EDGES * 32 + 255) / 256, 256, 0, stream>>>(
        ei, ew, deg, xw, agg, N_EDGES);
    // 6) final activation + tanh
    final_kernel<<<(int)((NF + 255) / 256), 256, 0, stream>>>(
        h1, hr, agg, barma, out, (int)NF);
}